// ResidualSAGEBlock_53807350284435
// MI455X (gfx1250) — compile-verified
//
#include <hip/hip_runtime.h>

// ResidualSAGEBlock for MI455X (gfx1250, wave32).
// Memory-bound workload: scatter-mean over 1.6M edges dominates (≈1.7 GB of
// traffic at 23.3 TB/s). GEMMs (2x 100k*128*128) are tiny -> use the exact-
// precision FP32 WMMA path (v_wmma_f32_16x16x4_f32) and fuse BN stat
// accumulation into the GEMM epilogue to avoid an extra pass over h.

typedef __attribute__((ext_vector_type(2))) float v2f;
typedef __attribute__((ext_vector_type(8))) float v8f;

#define DFEAT 128
#define BN_EPS 1e-5f
#define WPAD 129   // LDS row stride (in v2f) to spread banks

__device__ __forceinline__ v8f wmma4(v2f a, v2f b, v8f c) {
  // D = A(16x4,f32) * B(4x16,f32) + C(16x16,f32)
  return __builtin_amdgcn_wmma_f32_16x16x4_f32(
      /*neg_a=*/false, a, /*neg_b=*/false, b,
      /*c_mod=*/(short)0, c, /*reuse_a=*/false, /*reuse_b=*/false);
}

// ---------------------------------------------------------------- zero fill
__global__ void sage_zero_kernel(float* __restrict__ p, size_t n) {
  size_t i = (size_t)blockIdx.x * blockDim.x + threadIdx.x;
  size_t stride = (size_t)gridDim.x * blockDim.x;
  for (; i < n; i += stride) p[i] = 0.0f;
}

// ------------------------------------------------------------- scatter-add
// One wave32 per edge: 32 lanes x float4 = 128 floats (full row).
__global__ void sage_scatter_kernel(const float* __restrict__ x,
                                    const int* __restrict__ ei,
                                    float* __restrict__ agg,
                                    float* __restrict__ cnt, int E) {
  int wave = (int)(((size_t)blockIdx.x * blockDim.x + threadIdx.x) >> 5);
  int lane = threadIdx.x & 31;
  if (wave >= E) return;
  int src = ei[wave];      // edge_index[0][e]
  int dst = ei[E + wave];  // edge_index[1][e]
  const float4* xs = (const float4*)(x + (size_t)src * DFEAT);
  float4 v = xs[lane];
  float* a = agg + (size_t)dst * DFEAT + lane * 4;
  atomicAdd(a + 0, v.x);
  atomicAdd(a + 1, v.y);
  atomicAdd(a + 2, v.z);
  atomicAdd(a + 3, v.w);
  if (lane == 0) atomicAdd(cnt + dst, 1.0f);
}

// --------------------------------------------------------- agg /= max(cnt,1)
__global__ void sage_mean_kernel(float* __restrict__ agg,
                                 const float* __restrict__ cnt, int N) {
  size_t n4 = (size_t)N * (DFEAT / 4);
  size_t i = (size_t)blockIdx.x * blockDim.x + threadIdx.x;
  size_t stride = (size_t)gridDim.x * blockDim.x;
  float4* a4 = (float4*)agg;
  for (; i < n4; i += stride) {
    int row = (int)(i >> 5);  // 32 float4 per row
    float inv = 1.0f / fmaxf(cnt[row], 1.0f);
    float4 v = a4[i];
    v.x *= inv; v.y *= inv; v.z *= inv; v.w *= inv;
    a4[i] = v;
  }
}

// ------------------------------------------------ fused dual-GEMM + BN stats
// h = agg @ Wl^T + x @ Wr^T + bl ; also accumulate per-column sum / sumsq.
// Block: 256 threads (8 waves), 128 rows of output; each wave: 16 rows,
// all 8 column tiles, K swept 4 at a time with v_wmma_f32_16x16x4_f32.
__launch_bounds__(256) __global__
void sage_gemm_bn_kernel(const float* __restrict__ x,
                         const float* __restrict__ agg,
                         const float* __restrict__ Wl,
                         const float* __restrict__ Wr,
                         const float* __restrict__ bl,
                         float* __restrict__ h,
                         float* __restrict__ colsum,
                         float* __restrict__ colsq, int N) {
  // B-operand staging: sW[p*WPAD + j] = {W[j][2p], W[j][2p+1]}  (p = K pair)
  __shared__ v2f sWl[64 * WPAD];
  __shared__ v2f sWr[64 * WPAD];
  __shared__ float sSum[DFEAT];
  __shared__ float sSq[DFEAT];

  int tid = threadIdx.x;
  const v2f* Wl2 = (const v2f*)Wl;
  const v2f* Wr2 = (const v2f*)Wr;
  for (int idx = tid; idx < 64 * DFEAT; idx += 256) {
    int p = idx & 63;   // K pair index: coalesced 8B global reads along p
    int j = idx >> 6;   // output column
    sWl[p * WPAD + j] = Wl2[j * 64 + p];
    sWr[p * WPAD + j] = Wr2[j * 64 + p];
  }
  if (tid < DFEAT) { sSum[tid] = 0.0f; sSq[tid] = 0.0f; }
  __syncthreads();

  int waveId = tid >> 5;
  int lane = tid & 31;
  int half = lane >> 4;   // K sub-select per ISA f32 A/B layout
  int rc = lane & 15;     // A: row-in-tile ; B: col-in-tile
  int row0 = blockIdx.x * 128 + waveId * 16;
  int arow = row0 + rc;
  float amask = (arow < N) ? 1.0f : 0.0f;
  int crow = (arow < N) ? arow : 0;

  const v2f* ap = (const v2f*)(agg + (size_t)crow * DFEAT);
  const v2f* xp = (const v2f*)(x + (size_t)crow * DFEAT);

  v8f acc[8];
#pragma unroll
  for (int t = 0; t < 8; ++t) acc[t] = (v8f)0.0f;

#pragma unroll 4
  for (int k0 = 0; k0 < DFEAT; k0 += 4) {
    int kp = (k0 >> 1) + half;           // K pair for this half-wave
    v2f aL = ap[kp] * amask;             // A frag of agg (zero OOB rows)
    v2f aR = xp[kp] * amask;             // A frag of x
    const v2f* bLrow = &sWl[kp * WPAD];
    const v2f* bRrow = &sWr[kp * WPAD];
#pragma unroll
    for (int t = 0; t < 8; ++t) {
      v2f bL = bLrow[t * 16 + rc];
      v2f bR = bRrow[t * 16 + rc];
      acc[t] = wmma4(aL, bL, acc[t]);    // += agg @ Wl^T tile
      acc[t] = wmma4(aR, bR, acc[t]);    // += x   @ Wr^T tile
    }
  }

  // Epilogue: +bl, store h, per-column partial BN stats.
#pragma unroll
  for (int t = 0; t < 8; ++t) {
    int col = t * 16 + rc;
    float bv = bl[col];
    float s = 0.0f, sq = 0.0f;
#pragma unroll
    for (int r = 0; r < 8; ++r) {
      int row = row0 + r + half * 8;     // C layout: vgpr r -> M=r / M=r+8
      float v = acc[t][r] + bv;
      if (row < N) {
        h[(size_t)row * DFEAT + col] = v;
        s += v;
        sq += v * v;
      }
    }
    atomicAdd(&sSum[col], s);
    atomicAdd(&sSq[col], sq);
  }
  __syncthreads();
  if (tid < DFEAT) {
    atomicAdd(&colsum[tid], sSum[tid]);
    atomicAdd(&colsq[tid], sSq[tid]);
  }
}

// -------------------------------------------------------- BN scale / shift
__global__ void sage_bnfin_kernel(const float* __restrict__ colsum,
                                  const float* __restrict__ colsq,
                                  const float* __restrict__ gamma,
                                  const float* __restrict__ beta,
                                  float* __restrict__ scale,
                                  float* __restrict__ shift, float invN) {
  int j = threadIdx.x;
  float mean = colsum[j] * invN;
  float var = colsq[j] * invN - mean * mean;
  float sc = gamma[j] * rsqrtf(var + BN_EPS);
  scale[j] = sc;
  shift[j] = beta[j] - mean * sc;
}

// ------------------------------------------------- out = relu(h*sc + sh + x)
__global__ void sage_out_kernel(const float* __restrict__ h,
                                const float* __restrict__ x,
                                const float* __restrict__ scale,
                                const float* __restrict__ shift,
                                float* __restrict__ out, int N) {
  size_t n4 = (size_t)N * (DFEAT / 4);
  size_t i = (size_t)blockIdx.x * blockDim.x + threadIdx.x;
  size_t stride = (size_t)gridDim.x * blockDim.x;
  const float4* h4 = (const float4*)h;
  const float4* x4 = (const float4*)x;
  const float4* s4 = (const float4*)scale;
  const float4* b4 = (const float4*)shift;
  float4* o4 = (float4*)out;
  for (; i < n4; i += stride) {
    int c4 = (int)(i & 31);
    float4 hv = h4[i], xv = x4[i], sc = s4[c4], sh = b4[c4];
    float4 r;
    r.x = fmaxf(hv.x * sc.x + sh.x + xv.x, 0.0f);
    r.y = fmaxf(hv.y * sc.y + sh.y + xv.y, 0.0f);
    r.z = fmaxf(hv.z * sc.z + sh.z + xv.z, 0.0f);
    r.w = fmaxf(hv.w * sc.w + sh.w + xv.w, 0.0f);
    o4[i] = r;
  }
}

extern "C" void kernel_launch(void* const* d_in, const int* in_sizes, int n_in,
                              void* d_out, int out_size, void* d_ws,
                              size_t ws_size, hipStream_t stream) {
  const float* x     = (const float*)d_in[0];
  const int*   ei    = (const int*)d_in[1];
  const float* Wl    = (const float*)d_in[2];
  const float* bl    = (const float*)d_in[3];
  const float* Wr    = (const float*)d_in[4];
  const float* gamma = (const float*)d_in[5];
  const float* beta  = (const float*)d_in[6];
  float* out = (float*)d_out;

  int N = in_sizes[0] / DFEAT;
  int E = in_sizes[1] / 2;

  // Workspace layout (fp32): agg[N*D] | h[N*D] | cnt[N] | colsum[D] |
  //                          colsq[D] | scale[D] | shift[D]
  float* agg    = (float*)d_ws;
  float* h      = agg + (size_t)N * DFEAT;
  float* cnt    = h + (size_t)N * DFEAT;
  float* colsum = cnt + N;
  float* colsq  = colsum + DFEAT;
  float* scale  = colsq + DFEAT;
  float* shift  = scale + DFEAT;

  // 1) zero accumulators (every call: graph replays don't re-init ws)
  sage_zero_kernel<<<2048, 256, 0, stream>>>(agg, (size_t)N * DFEAT);
  sage_zero_kernel<<<512, 256, 0, stream>>>(cnt, (size_t)N + 2 * DFEAT);

  // 2) scatter-add messages (1 wave per edge)
  sage_scatter_kernel<<<(E + 7) / 8, 256, 0, stream>>>(x, ei, agg, cnt, E);

  // 3) mean aggregation
  sage_mean_kernel<<<2048, 256, 0, stream>>>(agg, cnt, N);

  // 4) fused dual WMMA GEMM + bias + BN stat accumulation
  sage_gemm_bn_kernel<<<(N + 127) / 128, 256, 0, stream>>>(
      x, agg, Wl, Wr, bl, h, colsum, colsq, N);

  // 5) BN scale/shift
  sage_bnfin_kernel<<<1, DFEAT, 0, stream>>>(colsum, colsq, gamma, beta,
                                             scale, shift, 1.0f / (float)N);

  // 6) residual + ReLU
  sage_out_kernel<<<2048, 256, 0, stream>>>(h, x, scale, shift, out, N);
}